// GNN_5488968204992
// MI455X (gfx1250) — compile-verified
//
#include <hip/hip_runtime.h>
#include <math.h>

// ---------------------------------------------------------------------------
// GNN: h = adj @ tanh(x@W1+b1); h = adj @ tanh(h@W2+b2); out = sum_n(h) @ W3 + b3
// B=64, N=1024, hidden=10 (padded to 16), out=2.
// Memory-bound on streaming adj (2 x 256MB @ 23.3 TB/s ~ 22us floor);
// fp32 WMMA (v_wmma_f32_16x16x4_f32) keeps full reference precision.
// ---------------------------------------------------------------------------

#define GNN_B   64
#define GNN_N   1024
#define GNN_DP  16      // padded feature dim (real = 10)

typedef __attribute__((ext_vector_type(2))) float v2f;
typedef __attribute__((ext_vector_type(8))) float v8f;

// ---- Kernel 1: m1^T[b][d][m] = tanh(x[b][m][:] @ W1 + b1), d padded to 16 ----
__global__ __launch_bounds__(256) void gnn_mlp1(
    const float* __restrict__ x, const float* __restrict__ W1,
    const float* __restrict__ b1, float* __restrict__ mt)
{
    const int idx = blockIdx.x * 256 + threadIdx.x;   // 0 .. B*N-1
    const int b = idx >> 10;
    const int m = idx & (GNN_N - 1);
    const float* xp = x + (size_t)idx * 3;
    const float x0 = xp[0], x1 = xp[1], x2 = xp[2];
    float* o = mt + (size_t)b * GNN_DP * GNN_N + m;   // stride N across d
#pragma unroll
    for (int d = 0; d < 10; ++d) {
        float v = x0 * W1[0 * 10 + d] + x1 * W1[1 * 10 + d] + x2 * W1[2 * 10 + d] + b1[d];
        o[d * GNN_N] = tanhf(v);
    }
#pragma unroll
    for (int d = 10; d < GNN_DP; ++d) o[d * GNN_N] = 0.0f;
}

// ---- Kernel 2: h[b][n][d] = sum_m adj[b][n][m] * mt[b][d][m]  (fp32 WMMA) ----
// Block = 8 waves = 128 output rows of one batch; mt[b] (64KB) staged in LDS.
// grid = B * (N/128) = 512 blocks.
__global__ __launch_bounds__(256) void gnn_agg_wmma(
    const float* __restrict__ adj, const float* __restrict__ mt,
    float* __restrict__ hout)
{
    __shared__ __align__(16) float lds_m[GNN_DP * GNN_N];   // 64 KB

    const int b      = blockIdx.x >> 3;    // 8 blocks per batch
    const int blk_in = blockIdx.x & 7;

    // cooperative fill of LDS with mt[b] : 4096 float4 / 256 threads
    {
        const float4* src = (const float4*)(mt + (size_t)b * GNN_DP * GNN_N);
        float4* dst = (float4*)lds_m;
#pragma unroll
        for (int i = 0; i < 16; ++i)
            dst[threadIdx.x + i * 256] = src[threadIdx.x + i * 256];
    }
    __syncthreads();

    const int wave = threadIdx.x >> 5;
    const int lane = threadIdx.x & 31;
    const int half = lane >> 4;            // 0: lanes 0-15, 1: lanes 16-31
    const int l16  = lane & 15;

    const int row0 = blk_in * 128 + wave * 16;          // tile base row
    // A operand (16x4 f32): row = l16, K = 2*half + vgpr
    const float* arow = adj + ((size_t)b * GNN_N + row0 + l16) * GNN_N + half * 2;
    // B operand (4x16 f32): N-col = l16 (feature d), K = 2*half + vgpr
    const float* brow = lds_m + l16 * GNN_N + half * 2;

    v8f acc = {0.f, 0.f, 0.f, 0.f, 0.f, 0.f, 0.f, 0.f};

    for (int k0 = 0; k0 < GNN_N; k0 += 64) {
        if (k0 < GNN_N - 256)
            __builtin_prefetch(arow + k0 + 256, 0, 1);  // global_prefetch ahead of the adj stream
#pragma unroll
        for (int k = 0; k < 64; k += 4) {
            v2f a = *(const v2f*)(arow + k0 + k);
            v2f bv = *(const v2f*)(brow + k0 + k);
            acc = __builtin_amdgcn_wmma_f32_16x16x4_f32(
                false, a, false, bv, (short)0, acc, false, false);
        }
    }

    // C layout: vgpr v -> M = v + 8*half, N-col = l16
    float* hp = hout + (((size_t)b * GNN_N + row0) * GNN_DP);
#pragma unroll
    for (int v = 0; v < 8; ++v) {
        const int mrow = v + half * 8;
        hp[mrow * GNN_DP + l16] = acc[v];
    }
}

// ---- Kernel 3: m2^T[b][d2][m] = tanh(h[b][m][:] @ W2 + b2), padded to 16 ----
__global__ __launch_bounds__(256) void gnn_mlp2(
    const float* __restrict__ h, const float* __restrict__ W2,
    const float* __restrict__ b2, float* __restrict__ mt)
{
    const int idx = blockIdx.x * 256 + threadIdx.x;
    const int b = idx >> 10;
    const int m = idx & (GNN_N - 1);
    const float* hp = h + (size_t)idx * GNN_DP;
    float hv[10];
#pragma unroll
    for (int d = 0; d < 10; ++d) hv[d] = hp[d];
    float* o = mt + (size_t)b * GNN_DP * GNN_N + m;
#pragma unroll
    for (int d2 = 0; d2 < 10; ++d2) {
        float v = b2[d2];
#pragma unroll
        for (int d = 0; d < 10; ++d) v += hv[d] * W2[d * 10 + d2];
        o[d2 * GNN_N] = tanhf(v);
    }
#pragma unroll
    for (int d2 = 10; d2 < GNN_DP; ++d2) o[d2 * GNN_N] = 0.0f;
}

// ---- Kernel 4: pooled[b] = sum_n h[b][n][:]; out[b] = pooled @ W3 + b3 ----
// One block per batch, deterministic LDS tree reduction.
__global__ __launch_bounds__(256) void gnn_reduce(
    const float* __restrict__ h, const float* __restrict__ W3,
    const float* __restrict__ b3, float* __restrict__ out)
{
    __shared__ float red[256];
    const int b = blockIdx.x;
    const int d = threadIdx.x & 15;
    const int g = threadIdx.x >> 4;        // 16 row-groups
    float s = 0.0f;
    const float* hb = h + (size_t)b * GNN_N * GNN_DP;
    for (int n = g; n < GNN_N; n += 16)
        s += hb[n * GNN_DP + d];
    red[threadIdx.x] = s;
    __syncthreads();
#pragma unroll
    for (int st = 128; st >= 16; st >>= 1) {
        if (threadIdx.x < st) red[threadIdx.x] += red[threadIdx.x + st];
        __syncthreads();
    }
    if (threadIdx.x < 2) {
        float o = b3[threadIdx.x];
#pragma unroll
        for (int dd = 0; dd < 10; ++dd) o += red[dd] * W3[dd * 2 + threadIdx.x];
        out[b * 2 + threadIdx.x] = o;
    }
}

// ---------------------------------------------------------------------------
extern "C" void kernel_launch(void* const* d_in, const int* in_sizes, int n_in,
                              void* d_out, int out_size, void* d_ws, size_t ws_size,
                              hipStream_t stream) {
    const float* x   = (const float*)d_in[0];
    const float* adj = (const float*)d_in[1];
    const float* W1  = (const float*)d_in[2];
    const float* b1  = (const float*)d_in[3];
    const float* W2  = (const float*)d_in[4];
    const float* b2  = (const float*)d_in[5];
    const float* W3  = (const float*)d_in[6];
    const float* b3  = (const float*)d_in[7];
    float* out = (float*)d_out;

    // workspace: mt (B*16*1024 f32 = 4MB) | h (B*1024*16 f32 = 4MB), ping-ponged
    float* mt   = (float*)d_ws;
    float* hbuf = (float*)((char*)d_ws + ((size_t)GNN_B * GNN_DP * GNN_N * sizeof(float)));

    const int node_blocks = (GNN_B * GNN_N) / 256;      // 256
    const int agg_blocks  = GNN_B * (GNN_N / 128);      // 512

    gnn_mlp1<<<node_blocks, 256, 0, stream>>>(x, W1, b1, mt);
    gnn_agg_wmma<<<agg_blocks, 256, 0, stream>>>(adj, mt, hbuf);
    gnn_mlp2<<<node_blocks, 256, 0, stream>>>(hbuf, W2, b2, mt);
    gnn_agg_wmma<<<agg_blocks, 256, 0, stream>>>(adj, mt, hbuf);
    gnn_reduce<<<GNN_B, 256, 0, stream>>>(hbuf, W3, b3, out);
}